// jslab_Ue_Unio_25881472925757
// MI455X (gfx1250) — compile-verified
//
#include <hip/hip_runtime.h>
#include <stdint.h>

#define NT   512          // one workgroup: 16 wave32s on one WGP
#define NMAX 2048         // max forcing length held in LDS (n = 1440 here)
#define RHO  1000.0f

typedef unsigned int u32;
typedef __attribute__((ext_vector_type(4))) unsigned int v4u;
typedef __attribute__((ext_vector_type(4))) int          v4i;
typedef __attribute__((ext_vector_type(8))) int          v8i;

#if defined(__has_builtin)
#if __has_builtin(__builtin_amdgcn_tensor_load_to_lds)
#define HAVE_TDM 1
#endif
#endif
#ifndef HAVE_TDM
#define HAVE_TDM 0
#endif

__device__ __forceinline__ void wait_tensor0() {
#if defined(__has_builtin) && __has_builtin(__builtin_amdgcn_s_wait_tensorcnt)
  __builtin_amdgcn_s_wait_tensorcnt(0);
#else
  asm volatile("s_wait_tensorcnt 0x0" ::: "memory");
#endif
}

#if HAVE_TDM
// 1-D contiguous tile load: n 4-byte elements from gptr -> LDS offset lds_off.
// D# layout per CDNA5 ISA ch.8 (group0: count/lds/global/type, group1: dims).
__device__ __forceinline__ void tdm_load_1d(const void* gptr, u32 lds_off, int n_elems) {
  uint64_t ga = (uint64_t)(uintptr_t)gptr;
  u32 n = (u32)n_elems;
  v4u g0;
  g0.x = 1u;                                     // count=1, user descriptor
  g0.y = lds_off;                                // lds_addr (bytes)
  g0.z = (u32)(ga & 0xFFFFFFFFu);                // global_addr[31:0]
  g0.w = (u32)((ga >> 32) & 0x01FFFFFFu)         // global_addr[56:32]
       | (2u << 30);                             // type = 2 ("image")
  v8i g1;
  g1[0] = (int)0x00020000u;                      // data_size=2 -> 4 bytes; no flags
  g1[1] = (int)((n & 0xFFFFu) << 16);            // tensor_dim0[15:0]  @bits 63:48
  g1[2] = (int)(((n >> 16) & 0xFFFFu)            // tensor_dim0[31:16] @bits 79:64
       | (1u << 16));                            // tensor_dim1 = 1
  g1[3] = (int)((n & 0xFFFFu) << 16);            // tile_dim0 = n      @bits 127:112
  g1[4] = (int)1;                                // tile_dim1 = 1, tile_dim2 = 0
  g1[5] = (int)n;                                // tensor_dim0_stride low
  g1[6] = 0;
  g1[7] = 0;
  v4i z4 = {0, 0, 0, 0};
#if __has_include(<hip/amd_detail/amd_gfx1250_TDM.h>)
  v8i z8 = {0, 0, 0, 0, 0, 0, 0, 0};
  __builtin_amdgcn_tensor_load_to_lds(g0, g1, z4, z4, z8, 0);   // clang-23 arity
#else
  __builtin_amdgcn_tensor_load_to_lds(g0, g1, z4, z4, 0);       // ROCm 7.2 arity
#endif
}
#endif

__global__ __launch_bounds__(NT) void jslab_unio_kernel(
    const float* __restrict__ pk,
    const float* __restrict__ TAx,
    const float* __restrict__ TAy,
    const float* __restrict__ fcP,
    const int* __restrict__ t0P, const int* __restrict__ t1P,
    const int* __restrict__ dtP, const int* __restrict__ dtfP,
    float* __restrict__ out, int n)
{
  // LDS: 6 arrays of NMAX + 4 scan arrays of NT = 14336 floats = 56 KB (<320 KB/WGP)
  __shared__ float smem[6 * NMAX + 4 * NT];
  float* sTx = smem;                 // raw TAx   (reused as sUr after Ce phase)
  float* sTy = smem + NMAX;          // raw TAy   (reused as sVr after Ce phase)
  float* sCr = smem + 2 * NMAX;      // Re(Ce) == Ue
  float* sCi = smem + 3 * NMAX;      // Im(Ce) == Ve
  float* sDr = smem + 4 * NMAX;      // Re(dCe/dt)
  float* sDi = smem + 5 * NMAX;      // Im(dCe/dt)
  float* mR  = smem + 6 * NMAX;      // scan: multiplier (complex)
  float* mI  = mR + NT;
  float* oR  = mI + NT;              // scan: offset (complex)
  float* oI  = oR + NT;
  float* sUr = sTx;                  // trajectory after each outer step
  float* sVr = sTy;

  const int tid = threadIdx.x;

  // ---- Stage forcing into LDS via the Tensor Data Mover (wave 0 issues) ----
#if HAVE_TDM
  if (tid < 32) {
    if (tid == 0) {
      tdm_load_1d(TAx, (u32)(uintptr_t)(void*)sTx, n);
      tdm_load_1d(TAy, (u32)(uintptr_t)(void*)sTy, n);
    }
    wait_tensor0();
  }
#else
  for (int i = tid; i < n; i += NT) { sTx[i] = TAx[i]; sTy[i] = TAy[i]; }
#endif
  __syncthreads();

  // ---- Scalars (device-resident; graph capture forbids host readback) ----
  const float K0  = expf(pk[0]);
  const float K1  = expf(pk[1]);
  const float fc  = fcP[0];
  const int   dt  = dtP[0];
  const int   dtF = dtfP[0];
  const int   nsub = dtF / dt;
  const int   Nf   = (t1P[0] - t0P[0]) / dtF;
  const float dtf  = (float)dt;

  // ---- Ekman equilibrium current: Ce = K0*(T)/(K1 + i fc), T = (TAx+iTAy)/RHO ----
  const float den = K1 * K1 + fc * fc;
  for (int i = tid; i < n; i += NT) {
    float Tr = sTx[i] * (1.0f / RHO);
    float Ti = sTy[i] * (1.0f / RHO);
    sCr[i] = K0 * (Tr * K1 + Ti * fc) / den;
    sCi[i] = K0 * (Ti * K1 - Tr * fc) / den;
  }
  __syncthreads();

  // ---- dCe/dt = jnp.gradient(Ce, dt): central interior, one-sided edges ----
  const float invdt  = 1.0f / dtf;
  const float inv2dt = 0.5f / dtf;
  for (int i = tid; i < n; i += NT) {
    if (i == 0) {
      sDr[0] = (sCr[1] - sCr[0]) * invdt;
      sDi[0] = (sCi[1] - sCi[0]) * invdt;
    } else if (i == n - 1) {
      sDr[i] = (sCr[i] - sCr[i - 1]) * invdt;
      sDi[i] = (sCi[i] - sCi[i - 1]) * invdt;
    } else {
      sDr[i] = (sCr[i + 1] - sCr[i - 1]) * inv2dt;
      sDi[i] = (sCi[i + 1] - sCi[i - 1]) * inv2dt;
    }
  }
  __syncthreads();

  // ---- Pass 1: per-thread chunk -> affine transform (H, S): w_out = H*w_in + S ----
  // H tracked as the homogeneous solution (start h=1+0i), S from w=0.
  const int chunk = (Nf + NT - 1) / NT;
  const int s = tid * chunk;
  const int e = (s + chunk < Nf) ? (s + chunk) : Nf;

  float ur = 0.0f, vr = 0.0f;        // forced solution from 0
  float hr = 1.0f, hi = 0.0f;        // homogeneous solution from 1
  const float invnsub = 1.0f / (float)nsub;
  for (int m = s; m < e; ++m) {
    const int msup = (m + 1 >= n) ? (n - 1) : (m + 1);
    const float D0r = sDr[m], D1r = sDr[msup];
    const float D0i = sDi[m], D1i = sDi[msup];
    for (int j = 0; j < nsub; ++j) {
      const float aa  = (float)j * invnsub;
      const float dUe = (1.0f - aa) * D0r + aa * D1r;
      const float dVe = (1.0f - aa) * D0i + aa * D1i;
      const float du  = -K1 * ur + fc * vr - dUe;
      const float dv  = -K1 * vr - fc * ur - dVe;
      const float dhr = -K1 * hr + fc * hi;
      const float dhi = -K1 * hi - fc * hr;
      ur += dtf * du;  vr += dtf * dv;
      hr += dtf * dhr; hi += dtf * dhi;
    }
  }
  mR[tid] = hr; mI[tid] = hi; oR[tid] = ur; oI[tid] = vr;   // idle threads: identity(1,0)+0
  __syncthreads();

  // ---- Inclusive Hillis-Steele scan over affine pairs (9 rounds) ----
  for (int off = 1; off < NT; off <<= 1) {
    float pmR = 0.f, pmI = 0.f, poR = 0.f, poI = 0.f;
    const float cmR = mR[tid], cmI = mI[tid], coR = oR[tid], coI = oI[tid];
    if (tid >= off) {
      pmR = mR[tid - off]; pmI = mI[tid - off];
      poR = oR[tid - off]; poI = oI[tid - off];
    }
    __syncthreads();
    if (tid >= off) {
      // compose: prev then cur -> (Hc*Hp, Hc*Sp + Sc), complex arithmetic
      mR[tid] = cmR * pmR - cmI * pmI;
      mI[tid] = cmR * pmI + cmI * pmR;
      oR[tid] = cmR * poR - cmI * poI + coR;
      oI[tid] = cmR * poI + cmI * poR + coI;
    }
    __syncthreads();
  }

  // ---- Pass 2: replay chunk from scanned entry state, record per-outer-step w ----
  float wr = 0.0f, wi = 0.0f;
  if (tid > 0) { wr = oR[tid - 1]; wi = oI[tid - 1]; }   // exclusive prefix (w0 = 0)
  for (int m = s; m < e; ++m) {
    const int msup = (m + 1 >= n) ? (n - 1) : (m + 1);
    const float D0r = sDr[m], D1r = sDr[msup];
    const float D0i = sDi[m], D1i = sDi[msup];
    for (int j = 0; j < nsub; ++j) {
      const float aa  = (float)j * invnsub;
      const float dUe = (1.0f - aa) * D0r + aa * D1r;
      const float dVe = (1.0f - aa) * D0i + aa * D1i;
      const float du  = -K1 * wr + fc * wi - dUe;
      const float dv  = -K1 * wi - fc * wr - dVe;
      wr += dtf * du; wi += dtf * dv;
    }
    sUr[m] = wr; sVr[m] = wi;
  }
  __syncthreads();

  // ---- Output: (Ue + Unio, Ve + Vnio); Unio = trajectory shifted right by one ----
  for (int m = tid; m < Nf; m += NT) {
    const float un = (m == 0) ? 0.0f : sUr[m - 1];
    const float vn = (m == 0) ? 0.0f : sVr[m - 1];
    out[m]      = sCr[m] + un;
    out[Nf + m] = sCi[m] + vn;
  }
}

extern "C" void kernel_launch(void* const* d_in, const int* in_sizes, int n_in,
                              void* d_out, int out_size, void* d_ws, size_t ws_size,
                              hipStream_t stream) {
  const float* pk  = (const float*)d_in[0];
  const float* TAx = (const float*)d_in[1];
  const float* TAy = (const float*)d_in[2];
  const float* fc  = (const float*)d_in[3];
  const int*   t0  = (const int*)d_in[4];
  const int*   t1  = (const int*)d_in[5];
  const int*   dt  = (const int*)d_in[6];
  const int*   dtf = (const int*)d_in[7];
  const int n = in_sizes[1];   // forcing length (1440)

  jslab_unio_kernel<<<dim3(1), dim3(NT), 0, stream>>>(
      pk, TAx, TAy, fc, t0, t1, dt, dtf, (float*)d_out, n);
}